// GlobalAttention_24369644437786
// MI455X (gfx1250) — compile-verified
//
#include <hip/hip_runtime.h>
#include <hip/hip_bf16.h>
#include <math.h>
#include <stdint.h>

// ---------------------------------------------------------------------------
// CDNA5 (gfx1250) implementation: windowed attention + separable conv, fp32.
// GEMMs: V_WMMA_F32_16X16X4_F32 (wave32). LDS tile staging: CDNA5 async
// global->LDS copies (GLOBAL_LOAD_ASYNC_TO_LDS_*, ASYNCcnt), bypassing VGPRs.
//
// Assumed WMMA f32 operand layouts (per CDNA5 ISA):
//   C/D 16x16: vgpr r, lane l -> C[M = r + 8*(l>=16)][N = l&15]
//   A   16x4 : lane l (m=l&15): {A[m][k0+koff], A[m][k0+koff+1]}, koff=(l<16?0:2)
//   B   4x16 : lane l (n=l&15): {B[k0+koff][n], B[k0+koff+1][n]}
// ---------------------------------------------------------------------------

typedef float v2f __attribute__((ext_vector_type(2)));
typedef float v8f __attribute__((ext_vector_type(8)));

#define NB      2
#define CQ      64
#define CV      256
#define IMG     256
#define PIX     65536            // IMG*IMG
#define WS_     8
#define HEADS_  8
#define NTOK    64               // WS*WS
#define DQ      8                // CQ/HEADS
#define DV      32               // CV/HEADS
#define SCALE   0.35355339059327373f   // (QD/HEADS)^-0.5 = 8^-0.5

__device__ __forceinline__ v8f wmma_f32(v2f a, v2f b, v8f c) {
  // 8 args: (neg_a, A, neg_b, B, c_mod, C, reuse_a, reuse_b)
  return __builtin_amdgcn_wmma_f32_16x16x4_f32(false, a, false, b, (short)0, c,
                                               false, false);
}

// --- CDNA5 async global->LDS copies (tracked by ASYNCcnt) -------------------
__device__ __forceinline__ void async_g2l_b32(unsigned lds_off, const void* g) {
  asm volatile("global_load_async_to_lds_b32 %0, %1, off"
               :: "v"(lds_off), "v"((unsigned long long)(uintptr_t)g)
               : "memory");
}
__device__ __forceinline__ void async_g2l_b128(unsigned lds_off, const void* g) {
  asm volatile("global_load_async_to_lds_b128 %0, %1, off"
               :: "v"(lds_off), "v"((unsigned long long)(uintptr_t)g)
               : "memory");
}
__device__ __forceinline__ void wait_async0() {
  asm volatile("s_wait_asynccnt 0" ::: "memory");
}

// ---------------------------------------------------------------------------
// Kernel 1: qp = Wq * q ; kp = Wk * qp    (fused; qp tile staged in LDS)
// Grid: 8192 blocks of 128 threads (4 waves). Tile = 16 pixels.
// Single shared allocation so async-LDS byte offsets are known (base 0):
//   [0, 4096)    q_s  [c][j]  stride 16
//   [4096, 8192) qp_s [o][j]  stride 16
// ---------------------------------------------------------------------------
__global__ __launch_bounds__(128) void qk_proj_kernel(
    const float* __restrict__ q, const float* __restrict__ Wq,
    const float* __restrict__ Wk, float* __restrict__ qp,
    float* __restrict__ kp) {
  __shared__ float sm[2 * CQ * 16];
  float* q_s  = sm;
  float* qp_s = sm + CQ * 16;

  const int tid  = threadIdx.x;
  const int wave = tid >> 5;
  const int lane = tid & 31;
  const int lm   = lane & 15;
  const int koff = (lane < 16) ? 0 : 2;
  const int hi8  = (lane < 16) ? 0 : 8;

  const int P0 = blockIdx.x * 16;
  const int b  = P0 >> 16;          // 65536 pixels per batch image
  const int p0 = P0 & (PIX - 1);
  const float* qb = q + (b * CQ) * PIX;

  // async-stage q tile (64 x 16 floats = 256 x b128 chunks) into LDS
  for (int ci = tid; ci < CQ * 4; ci += 128) {
    int c = ci >> 2, jq = (ci & 3) << 2;
    async_g2l_b128((unsigned)(c * 16 + jq) * 4u, qb + c * PIX + p0 + jq);
  }
  wait_async0();
  __syncthreads();

  // qp tile: rows [16*wave, 16*wave+16), K = 64
  v8f acc = {};
#pragma unroll
  for (int k0 = 0; k0 < CQ; k0 += 4) {
    const int ka = k0 + koff;
    v2f a, bb;
    a.x  = Wq[(wave * 16 + lm) * CQ + ka];
    a.y  = Wq[(wave * 16 + lm) * CQ + ka + 1];
    bb.x = q_s[ka * 16 + lm];
    bb.y = q_s[(ka + 1) * 16 + lm];
    acc = wmma_f32(a, bb, acc);
  }
#pragma unroll
  for (int r = 0; r < 8; ++r) {
    int o = wave * 16 + r + hi8;
    qp_s[o * 16 + lm] = acc[r];
    qp[(b * CQ + o) * PIX + p0 + lm] = acc[r];
  }
  __syncthreads();

  // kp tile: Wk * qp_s
  v8f acc2 = {};
#pragma unroll
  for (int k0 = 0; k0 < CQ; k0 += 4) {
    const int ka = k0 + koff;
    v2f a, bb;
    a.x  = Wk[(wave * 16 + lm) * CQ + ka];
    a.y  = Wk[(wave * 16 + lm) * CQ + ka + 1];
    bb.x = qp_s[ka * 16 + lm];
    bb.y = qp_s[(ka + 1) * 16 + lm];
    acc2 = wmma_f32(a, bb, acc2);
  }
#pragma unroll
  for (int r = 0; r < 8; ++r) {
    int o = wave * 16 + r + hi8;
    kp[(b * CQ + o) * PIX + p0 + lm] = acc2[r];
  }
}

// ---------------------------------------------------------------------------
// Kernel 2/5: Y[o,p] = sum_c W[o,c] * X[c,p], C = K = 256, per 16-pixel tile.
// Grid: 8192 blocks of 256 threads (8 waves). Wave w: rows [32w, 32w+32).
// X tile async-staged into LDS; stride 20 floats (80 B) -> b128 chunks stay
// 16B-aligned in LDS and row reads stay bank-conflict-free.
// ---------------------------------------------------------------------------
#define XSTR 20

__global__ __launch_bounds__(256) void conv1x1_256_kernel(
    const float* __restrict__ X, const float* __restrict__ W,
    float* __restrict__ Y) {
  __shared__ float x_s[CV * XSTR];

  const int tid  = threadIdx.x;
  const int wave = tid >> 5;
  const int lane = tid & 31;
  const int lm   = lane & 15;
  const int koff = (lane < 16) ? 0 : 2;
  const int hi8  = (lane < 16) ? 0 : 8;

  const int P0 = blockIdx.x * 16;
  const int b  = P0 >> 16;
  const int p0 = P0 & (PIX - 1);
  const float* xb = X + (b * CV) * PIX;

  // async-stage X tile (256 x 16 floats = 1024 x b128 chunks)
  for (int ci = tid; ci < CV * 4; ci += 256) {
    int c = ci >> 2, jq = (ci & 3) << 2;
    async_g2l_b128((unsigned)(c * XSTR + jq) * 4u, xb + c * PIX + p0 + jq);
  }
  wait_async0();
  __syncthreads();

#pragma unroll
  for (int t = 0; t < 2; ++t) {
    const int obase = wave * 32 + t * 16;
    // speculative prefetch of this tile's weight rows (global_prefetch_b8)
    __builtin_prefetch(&W[(obase + lm) * CV], 0, 1);
    v8f acc = {};
#pragma unroll 8
    for (int k0 = 0; k0 < CV; k0 += 4) {
      const int ka = k0 + koff;
      v2f a, bb;
      a.x  = W[(obase + lm) * CV + ka];
      a.y  = W[(obase + lm) * CV + ka + 1];
      bb.x = x_s[ka * XSTR + lm];
      bb.y = x_s[(ka + 1) * XSTR + lm];
      acc = wmma_f32(a, bb, acc);
    }
#pragma unroll
    for (int r = 0; r < 8; ++r) {
      int o = obase + r + hi8;
      Y[(b * CV + o) * PIX + p0 + lm] = acc[r];
    }
  }
}

// ---------------------------------------------------------------------------
// Kernel 3: windowed attention. 1 block = 1 window (8x8 = 64 tokens),
// 1 wave = 1 head. 228 KB dynamic LDS/block (CDNA5: 320 KB per WGP).
// q/k/v staged via async b32 gathers (dynamic LDS base offset 0, no statics).
// S = (q k^T)*scale + bias -> row softmax in LDS -> O = P v, scattered
// directly into unwindowed (b, C, H, W) layout.
// ---------------------------------------------------------------------------
#define VSTR  33
#define SSTR  65
#define WQK_F 512                       // 64 tok * 8 dims
#define WV_F  (NTOK * VSTR)             // 2112
#define WS_F  (NTOK * SSTR)             // 4160
#define PERW  (2 * WQK_F + WV_F + WS_F) // 7296 floats / wave

__global__ __launch_bounds__(256) void attn_kernel(
    const float* __restrict__ qp, const float* __restrict__ kp,
    const float* __restrict__ vp, const float* __restrict__ bias_table,
    const int* __restrict__ rel_index, float* __restrict__ out) {
  extern __shared__ float smem[];

  const int tid  = threadIdx.x;
  const int h    = tid >> 5;            // head = wave
  const int lane = tid & 31;
  const int lm   = lane & 15;
  const int koff = (lane < 16) ? 0 : 2;
  const int hi8  = (lane < 16) ? 0 : 8;

  float* q_s = smem + h * PERW;
  float* k_s = q_s + WQK_F;
  float* v_s = k_s + WQK_F;
  float* S_s = v_s + WV_F;
  const unsigned wbase = (unsigned)(h * PERW) * 4u;   // LDS byte offset

  const int wi = blockIdx.x;            // 2048 windows
  const int b  = wi >> 10;
  const int rm = wi & 1023;
  const int hh = rm >> 5, ww = rm & 31;
  const int pbase = hh * (WS_ * IMG) + ww * WS_;

  const float* qb = qp + (b * CQ) * PIX;
  const float* kb = kp + (b * CQ) * PIX;
  const float* vb = vp + (b * CV) * PIX;

  // async-stage this head's q/k (64x8) and v (64x32) tiles into LDS
  for (int i = lane; i < WQK_F; i += 32) {
    int t = i >> 3, d = i & 7;
    int p = pbase + ((t >> 3) << 8) + (t & 7);
    async_g2l_b32(wbase + 4u * (unsigned)i,           qb + (h * DQ + d) * PIX + p);
    async_g2l_b32(wbase + 4u * (unsigned)(WQK_F + i), kb + (h * DQ + d) * PIX + p);
  }
  for (int i = lane; i < NTOK * DV; i += 32) {
    int t = i >> 5, d = i & 31;
    int p = pbase + ((t >> 3) << 8) + (t & 7);
    async_g2l_b32(wbase + 4u * (unsigned)(2 * WQK_F + t * VSTR + d),
                  vb + (h * DV + d) * PIX + p);
  }
  wait_async0();
  __syncthreads();

  // S = q k^T * scale + bias   (M=64, N=64, K=8) -> 16 tiles x 2 WMMA
#pragma unroll
  for (int ti = 0; ti < 4; ++ti) {
#pragma unroll
    for (int tj = 0; tj < 4; ++tj) {
      v8f acc = {};
#pragma unroll
      for (int s = 0; s < 2; ++s) {
        const int ka = s * 4 + koff;
        v2f a, bb;
        a.x  = q_s[(ti * 16 + lm) * DQ + ka];
        a.y  = q_s[(ti * 16 + lm) * DQ + ka + 1];
        bb.x = k_s[(tj * 16 + lm) * DQ + ka];      // B[k][m] = k_s[m][k]
        bb.y = k_s[(tj * 16 + lm) * DQ + ka + 1];
        acc = wmma_f32(a, bb, acc);
      }
#pragma unroll
      for (int r = 0; r < 8; ++r) {
        int n = ti * 16 + r + hi8;
        int m = tj * 16 + lm;
        float bias = bias_table[rel_index[n * NTOK + m] * HEADS_ + h];
        S_s[n * SSTR + m] = acc[r] * SCALE + bias;
      }
    }
  }
  __syncthreads();

  // row softmax: 2 rows per lane
  for (int n = lane; n < NTOK; n += 32) {
    float* row = S_s + n * SSTR;
    float mx = -3.4e38f;
#pragma unroll
    for (int m = 0; m < NTOK; ++m) mx = fmaxf(mx, row[m]);
    float sum = 0.f;
#pragma unroll
    for (int m = 0; m < NTOK; ++m) {
      float e = __expf(row[m] - mx);
      row[m] = e;
      sum += e;
    }
    float inv = __frcp_rn(sum);
#pragma unroll
    for (int m = 0; m < NTOK; ++m) row[m] *= inv;
  }
  __syncthreads();

  // O = P v   (M=64, N=32, K=64) -> 8 tiles x 16 WMMA, scatter to (b,C,H,W)
  float* ob = out + (b * CV) * PIX;
#pragma unroll
  for (int ti = 0; ti < 4; ++ti) {
#pragma unroll
    for (int tj = 0; tj < 2; ++tj) {
      v8f acc = {};
#pragma unroll
      for (int ks = 0; ks < 16; ++ks) {
        const int ka = ks * 4 + koff;
        v2f a, bb;
        a.x  = S_s[(ti * 16 + lm) * SSTR + ka];
        a.y  = S_s[(ti * 16 + lm) * SSTR + ka + 1];
        bb.x = v_s[ka * VSTR + tj * 16 + lm];       // B[k][d] = v_s[k][d]
        bb.y = v_s[(ka + 1) * VSTR + tj * 16 + lm];
        acc = wmma_f32(a, bb, acc);
      }
#pragma unroll
      for (int r = 0; r < 8; ++r) {
        int n = ti * 16 + r + hi8;                  // token
        int d = tj * 16 + lm;                       // head dim
        int p = pbase + ((n >> 3) << 8) + (n & 7);
        ob[(h * DV + d) * PIX + p] = acc[r];
      }
    }
  }
}

// ---------------------------------------------------------------------------
// Kernel 4: reflect-pad(r/b=1) + depthwise 8x8 conv (zero pad 3) + BN(eval).
// Block = 256 thr = one 16x16 output tile of one (b, c) plane; 23x23 halo
// in LDS. Padded coord py in [0,256]; py==256 reflects to 254; outside -> 0.
// ---------------------------------------------------------------------------
__global__ __launch_bounds__(256) void dwconv_bn_kernel(
    const float* __restrict__ src, const float* __restrict__ dwk,
    const float* __restrict__ gamma, const float* __restrict__ beta,
    const float* __restrict__ mean, const float* __restrict__ var,
    float* __restrict__ dst) {
  __shared__ float in_s[23 * 24];
  __shared__ float wt[64];

  const int tid = threadIdx.x;
  const int bc  = blockIdx.z;           // b*256 + c
  const int c   = bc & (CV - 1);
  const float* plane = src + bc * PIX;

  if (tid < 64) wt[tid] = dwk[c * 64 + tid];

  const int y0 = blockIdx.y * 16 - 3;
  const int x0 = blockIdx.x * 16 - 3;
  for (int i = tid; i < 23 * 23; i += 256) {
    int iy = i / 23, ix = i % 23;
    int yy = y0 + iy, xx = x0 + ix;
    float v = 0.f;
    if (yy >= 0 && yy <= IMG && xx >= 0 && xx <= IMG) {
      int ry = (yy == IMG) ? IMG - 2 : yy;
      int rx = (xx == IMG) ? IMG - 2 : xx;
      v = plane[ry * IMG + rx];
    }
    in_s[iy * 24 + ix] = v;
  }
  __syncthreads();

  const int ty = tid >> 4, tx = tid & 15;
  float acc = 0.f;
#pragma unroll
  for (int i = 0; i < 8; ++i)
#pragma unroll
    for (int j = 0; j < 8; ++j)
      acc = fmaf(in_s[(ty + i) * 24 + (tx + j)], wt[i * 8 + j], acc);

  const float inv = gamma[c] * rsqrtf(var[c] + 1e-5f);
  acc = acc * inv + (beta[c] - mean[c] * inv);
  dst[bc * PIX + (blockIdx.y * 16 + ty) * IMG + blockIdx.x * 16 + tx] = acc;
}

// ---------------------------------------------------------------------------
extern "C" void kernel_launch(void* const* d_in, const int* in_sizes, int n_in,
                              void* d_out, int out_size, void* d_ws,
                              size_t ws_size, hipStream_t stream) {
  (void)in_sizes; (void)n_in; (void)out_size; (void)ws_size;

  const float* q          = (const float*)d_in[0];
  const float* v          = (const float*)d_in[1];
  const float* Wq         = (const float*)d_in[2];
  const float* Wk         = (const float*)d_in[3];
  const float* Wv         = (const float*)d_in[4];
  const float* bias_table = (const float*)d_in[5];
  const float* dw_kernel  = (const float*)d_in[6];
  const float* bn_gamma   = (const float*)d_in[7];
  const float* bn_beta    = (const float*)d_in[8];
  const float* bn_mean    = (const float*)d_in[9];
  const float* bn_var     = (const float*)d_in[10];
  const float* pw_kernel  = (const float*)d_in[11];
  const int*   rel_index  = (const int*)d_in[12];

  float* ws = (float*)d_ws;
  const size_t QP_N = (size_t)NB * CQ * PIX;   //  8.39 M floats
  const size_t VP_N = (size_t)NB * CV * PIX;   // 33.55 M floats
  float* qp   = ws;
  float* kp   = qp + QP_N;
  float* vp   = kp + QP_N;
  float* attn = vp + VP_N;
  float* dw   = vp;            // vp dead after attention -> reuse for dw out

  const int npix_tiles = (NB * PIX) / 16;      // 8192

  qk_proj_kernel<<<npix_tiles, 128, 0, stream>>>(q, Wq, Wk, qp, kp);
  conv1x1_256_kernel<<<npix_tiles, 256, 0, stream>>>(v, Wv, vp);

  const size_t shmem = (size_t)HEADS_ * PERW * sizeof(float);  // 233472 B
  attn_kernel<<<NB * 1024, 256, shmem, stream>>>(qp, kp, vp, bias_table,
                                                 rel_index, attn);

  dim3 g4(IMG / 16, IMG / 16, NB * CV);
  dwconv_bn_kernel<<<g4, 256, 0, stream>>>(attn, dw_kernel, bn_gamma, bn_beta,
                                           bn_mean, bn_var, dw);

  conv1x1_256_kernel<<<npix_tiles, 256, 0, stream>>>(dw, pw_kernel,
                                                     (float*)d_out);
}